// MultiHeadAttention_35794257445291
// MI455X (gfx1250) — compile-verified
//
#include <hip/hip_runtime.h>

#define D_MODEL 1024
#define NHEAD   16
#define DK      64
#define DV      64
#define BB      4
#define SQ      2048
#define LN_EPS  1e-5f

typedef __attribute__((ext_vector_type(2))) float v2f;
typedef __attribute__((ext_vector_type(8))) float v8f;

// ---------------------------------------------------------------------------
// wave32 reductions
// ---------------------------------------------------------------------------
__device__ __forceinline__ float wave_max(float v) {
    for (int off = 16; off > 0; off >>= 1)
        v = fmaxf(v, __shfl_xor(v, off, 32));
    return v;
}
__device__ __forceinline__ float wave_sum(float v) {
    for (int off = 16; off > 0; off >>= 1)
        v += __shfl_xor(v, off, 32);
    return v;
}

// f32 WMMA: D = A(16x4) * B(4x16) + C, one tile per wave.
__device__ __forceinline__ v8f wmma4(v2f a, v2f b, v8f c) {
    return __builtin_amdgcn_wmma_f32_16x16x4_f32(false, a, false, b,
                                                 (short)0, c, false, false);
}

// async 16B global->LDS copy (CDNA5 GLOBAL_LOAD_ASYNC_TO_LDS_B128, ASYNCcnt)
__device__ __forceinline__ void async_copy16(unsigned lds_off, const float* g) {
    asm volatile("global_load_async_to_lds_b128 %0, %1, off"
                 :: "v"(lds_off), "v"((unsigned long long)g) : "memory");
}
__device__ __forceinline__ void async_wait0() {
    asm volatile("s_wait_asynccnt 0x0" ::: "memory");
}

// ---------------------------------------------------------------------------
// 1) QKV projection: [B*S,1024] @ [1024,1024] -> head-major [B,H,S,64]
//    wave computes 16 rows x 64 cols (4 accumulators); block = 128 x 64.
// ---------------------------------------------------------------------------
__global__ void __launch_bounds__(256)
proj_kernel(const float* __restrict__ Qin, const float* __restrict__ Kin,
            const float* __restrict__ Vin, const float* __restrict__ WQ,
            const float* __restrict__ WK,  const float* __restrict__ WV,
            float* __restrict__ Qh, float* __restrict__ Kh, float* __restrict__ Vh)
{
    const int which = blockIdx.z;
    const float* A = (which == 0) ? Qin : (which == 1) ? Kin : Vin;
    const float* W = (which == 0) ? WQ  : (which == 1) ? WK  : WV;
    float*       O = (which == 0) ? Qh  : (which == 1) ? Kh  : Vh;

    const int lane = threadIdx.x & 31;
    const int wave = threadIdx.x >> 5;
    const int half = lane >> 4;
    const int lm   = lane & 15;
    const int row0 = blockIdx.y * 128 + wave * 16;   // token index b*S+s
    const int col0 = blockIdx.x * 64;                // h*64 base (head = blockIdx.x)

    const float* arow = A + (size_t)(row0 + lm) * D_MODEL;
    v8f acc[4] = {};
    for (int kc = 0; kc < D_MODEL; kc += 4) {
        const int ka = kc + 2 * half;
        v2f a;
        a.x = arow[ka];
        a.y = arow[ka + 1];
        #pragma unroll
        for (int t = 0; t < 4; ++t) {
            v2f b;
            b.x = W[(size_t)ka       * D_MODEL + col0 + t * 16 + lm];
            b.y = W[(size_t)(ka + 1) * D_MODEL + col0 + t * 16 + lm];
            acc[t] = wmma4(a, b, acc[t]);
        }
    }

    const int h = blockIdx.x;                        // 64-wide strip == one head
    #pragma unroll
    for (int t = 0; t < 4; ++t) {
        const int d = t * 16 + lm;
        #pragma unroll
        for (int r = 0; r < 8; ++r) {
            const int m  = row0 + r + 8 * half;      // b*S+s
            const int b_ = m >> 11;
            const int s  = m & (SQ - 1);
            O[(((size_t)b_ * NHEAD + h) * SQ + s) * DK + d] = acc[t][r];
        }
    }
}

// ---------------------------------------------------------------------------
// 2) scores = Q @ K^T / d_k**0.25, mask applied.
//    LDS-staged: Q tile 128x64, K tile 64x64 via async global->LDS DMA.
// ---------------------------------------------------------------------------
#define LP 65   // LDS row pitch (floats), padded to spread banks
__global__ void __launch_bounds__(256)
scores_kernel(const float* __restrict__ Qh, const float* __restrict__ Kh,
              const unsigned char* __restrict__ mask, float* __restrict__ attn)
{
    __shared__ float Ql[128 * LP];
    __shared__ float Kl[64 * LP];

    const int bh = blockIdx.z;                 // b*NHEAD + h
    const int b_ = bh >> 4;
    const float* Qp = Qh   + (size_t)bh * SQ * DK;
    const float* Kp = Kh   + (size_t)bh * SQ * DK;
    float*       Op = attn + (size_t)bh * SQ * SQ;

    const int tid   = threadIdx.x;
    const int qrow0 = blockIdx.y * 128;
    const int kcol0 = blockIdx.x * 64;

    // stage Q tile: 128 rows x 64 f32 = 2048 x 16B chunks (8 per thread)
    #pragma unroll
    for (int i = 0; i < 8; ++i) {
        const int ch  = tid + i * 256;
        const int row = ch >> 4;               // 16 chunks per row
        const int cc  = (ch & 15) * 4;         // float column
        async_copy16((unsigned)(size_t)&Ql[row * LP + cc],
                     Qp + (size_t)(qrow0 + row) * DK + cc);
    }
    // stage K tile: 64 rows x 64 f32 = 1024 x 16B chunks (4 per thread)
    #pragma unroll
    for (int i = 0; i < 4; ++i) {
        const int ch  = tid + i * 256;
        const int row = ch >> 4;
        const int cc  = (ch & 15) * 4;
        async_copy16((unsigned)(size_t)&Kl[row * LP + cc],
                     Kp + (size_t)(kcol0 + row) * DK + cc);
    }
    async_wait0();
    __syncthreads();

    const int lane = tid & 31;
    const int wave = tid >> 5;
    const int half = lane >> 4;
    const int lm   = lane & 15;
    const int arow = wave * 16 + lm;           // row within Q tile

    v8f acc[4] = {};
    for (int kc = 0; kc < DK; kc += 4) {
        const int ka = kc + 2 * half;
        v2f a;
        a.x = Ql[arow * LP + ka];
        a.y = Ql[arow * LP + ka + 1];
        #pragma unroll
        for (int t = 0; t < 4; ++t) {
            v2f b;
            b.x = Kl[(t * 16 + lm) * LP + ka];
            b.y = Kl[(t * 16 + lm) * LP + ka + 1];
            acc[t] = wmma4(a, b, acc[t]);
        }
    }

    const float inv_scale = 0.35355339059327373f;  // 1 / 64**0.25
    const unsigned char* mb = mask + (size_t)b_ * SQ * SQ;
    #pragma unroll
    for (int t = 0; t < 4; ++t) {
        const int k = kcol0 + t * 16 + lm;
        #pragma unroll
        for (int r = 0; r < 8; ++r) {
            const int q = qrow0 + wave * 16 + r + 8 * half;
            float v = acc[t][r] * inv_scale;
            if (mb[(size_t)q * SQ + k]) v = -1e9f;
            Op[(size_t)q * SQ + k] = v;
        }
    }
}

// ---------------------------------------------------------------------------
// 3) row softmax over 2048 elements, in place on attn output.
// ---------------------------------------------------------------------------
__global__ void __launch_bounds__(256)
softmax_kernel(float* __restrict__ attn)
{
    float* p = attn + (size_t)blockIdx.x * SQ;
    const int tid  = threadIdx.x;
    const int lane = tid & 31;
    const int wave = tid >> 5;
    __shared__ float sred[8];
    __shared__ float sbc;

    float vals[8];
    float m = -3.4e38f;
    #pragma unroll
    for (int i = 0; i < 8; ++i) {
        vals[i] = p[tid + i * 256];
        m = fmaxf(m, vals[i]);
    }
    m = wave_max(m);
    if (lane == 0) sred[wave] = m;
    __syncthreads();
    if (tid == 0) {
        float t = sred[0];
        for (int i = 1; i < 8; ++i) t = fmaxf(t, sred[i]);
        sbc = t;
    }
    __syncthreads();
    const float rowmax = sbc;

    float sum = 0.f;
    #pragma unroll
    for (int i = 0; i < 8; ++i) {
        vals[i] = __expf(vals[i] - rowmax);
        sum += vals[i];
    }
    sum = wave_sum(sum);
    __syncthreads();
    if (lane == 0) sred[wave] = sum;
    __syncthreads();
    if (tid == 0) {
        float t = 0.f;
        for (int i = 0; i < 8; ++i) t += sred[i];
        sbc = t;
    }
    __syncthreads();
    const float inv = 1.0f / sbc;
    #pragma unroll
    for (int i = 0; i < 8; ++i)
        p[tid + i * 256] = vals[i] * inv;
}

// ---------------------------------------------------------------------------
// 4) context = attn @ V  -> [B, S, H*64]; wave = 16 rows x all 64 cols.
// ---------------------------------------------------------------------------
__global__ void __launch_bounds__(256)
av_kernel(const float* __restrict__ attn, const float* __restrict__ Vh,
          float* __restrict__ ctx)
{
    const int bh = blockIdx.z;
    const int b_ = bh >> 4;
    const int h  = bh & 15;
    const float* Ap = attn + (size_t)bh * SQ * SQ;
    const float* Vp = Vh   + (size_t)bh * SQ * DV;

    const int lane = threadIdx.x & 31;
    const int wave = threadIdx.x >> 5;
    const int half = lane >> 4;
    const int lm   = lane & 15;
    const int row0 = blockIdx.y * 128 + wave * 16;  // q

    const float* arow = Ap + (size_t)(row0 + lm) * SQ;
    v8f acc[4] = {};
    for (int kc = 0; kc < SQ; kc += 4) {
        const int ka = kc + 2 * half;
        v2f a;
        a.x = arow[ka];
        a.y = arow[ka + 1];
        #pragma unroll
        for (int t = 0; t < 4; ++t) {
            v2f b;
            b.x = Vp[(size_t)ka       * DV + t * 16 + lm];
            b.y = Vp[(size_t)(ka + 1) * DV + t * 16 + lm];
            acc[t] = wmma4(a, b, acc[t]);
        }
    }
    #pragma unroll
    for (int t = 0; t < 4; ++t) {
        const int d = t * 16 + lm;
        #pragma unroll
        for (int r = 0; r < 8; ++r) {
            const int q = row0 + r + 8 * half;
            ctx[((size_t)b_ * SQ + q) * D_MODEL + h * DV + d] = acc[t][r];
        }
    }
}

// ---------------------------------------------------------------------------
// 5) out = ctx @ W_fc + residual(Q_input)  (pre-LayerNorm)
// ---------------------------------------------------------------------------
__global__ void __launch_bounds__(256)
outproj_kernel(const float* __restrict__ ctx, const float* __restrict__ Wfc,
               const float* __restrict__ Qin, float* __restrict__ out)
{
    const int lane = threadIdx.x & 31;
    const int wave = threadIdx.x >> 5;
    const int half = lane >> 4;
    const int lm   = lane & 15;
    const int row0 = blockIdx.y * 128 + wave * 16;
    const int col0 = blockIdx.x * 64;

    const float* arow = ctx + (size_t)(row0 + lm) * D_MODEL;
    v8f acc[4] = {};
    for (int kc = 0; kc < D_MODEL; kc += 4) {
        const int ka = kc + 2 * half;
        v2f a;
        a.x = arow[ka];
        a.y = arow[ka + 1];
        #pragma unroll
        for (int t = 0; t < 4; ++t) {
            v2f b;
            b.x = Wfc[(size_t)ka       * D_MODEL + col0 + t * 16 + lm];
            b.y = Wfc[(size_t)(ka + 1) * D_MODEL + col0 + t * 16 + lm];
            acc[t] = wmma4(a, b, acc[t]);
        }
    }
    #pragma unroll
    for (int t = 0; t < 4; ++t) {
        const int n = col0 + t * 16 + lm;
        #pragma unroll
        for (int r = 0; r < 8; ++r) {
            const int m = row0 + r + 8 * half;
            out[(size_t)m * D_MODEL + n] = acc[t][r] + Qin[(size_t)m * D_MODEL + n];
        }
    }
}

// ---------------------------------------------------------------------------
// 6) LayerNorm over last dim (1024), in place on d_out q-part.
// ---------------------------------------------------------------------------
__global__ void __launch_bounds__(256)
ln_kernel(float* __restrict__ out, const float* __restrict__ gamma,
          const float* __restrict__ beta)
{
    float* p = out + (size_t)blockIdx.x * D_MODEL;
    const int tid  = threadIdx.x;
    const int lane = tid & 31;
    const int wave = tid >> 5;
    __shared__ float sred[8];
    __shared__ float sbc;

    float vals[4];
    float sum = 0.f;
    #pragma unroll
    for (int i = 0; i < 4; ++i) {
        vals[i] = p[tid + i * 256];
        sum += vals[i];
    }
    sum = wave_sum(sum);
    if (lane == 0) sred[wave] = sum;
    __syncthreads();
    if (tid == 0) {
        float t = 0.f;
        for (int i = 0; i < 8; ++i) t += sred[i];
        sbc = t * (1.0f / D_MODEL);
    }
    __syncthreads();
    const float mu = sbc;

    float vs = 0.f;
    #pragma unroll
    for (int i = 0; i < 4; ++i) {
        const float d = vals[i] - mu;
        vs += d * d;
    }
    vs = wave_sum(vs);
    __syncthreads();
    if (lane == 0) sred[wave] = vs;
    __syncthreads();
    if (tid == 0) {
        float t = 0.f;
        for (int i = 0; i < 8; ++i) t += sred[i];
        sbc = rsqrtf(t * (1.0f / D_MODEL) + LN_EPS);
    }
    __syncthreads();
    const float rstd = sbc;
    #pragma unroll
    for (int i = 0; i < 4; ++i) {
        const int n = tid + i * 256;
        p[n] = (vals[i] - mu) * rstd * gamma[n] + beta[n];
    }
}

// ---------------------------------------------------------------------------
extern "C" void kernel_launch(void* const* d_in, const int* in_sizes, int n_in,
                              void* d_out, int out_size, void* d_ws, size_t ws_size,
                              hipStream_t stream)
{
    const float*         Qin  = (const float*)d_in[0];
    const float*         Kin  = (const float*)d_in[1];
    const float*         Vin  = (const float*)d_in[2];
    const unsigned char* mask = (const unsigned char*)d_in[3];
    const float*         WQ   = (const float*)d_in[4];
    const float*         WK   = (const float*)d_in[5];
    const float*         WV   = (const float*)d_in[6];
    const float*         Wfc  = (const float*)d_in[7];
    const float*         gam  = (const float*)d_in[8];
    const float*         bet  = (const float*)d_in[9];

    float* outq = (float*)d_out;                                // [B,S,1024]
    float* attn = outq + (size_t)BB * SQ * D_MODEL;             // [B,H,S,S]

    const size_t HEADSZ = (size_t)BB * NHEAD * SQ * DK;         // 8,388,608 floats
    float* Qh  = (float*)d_ws;
    float* Kh  = Qh + HEADSZ;
    float* Vh  = Kh + HEADSZ;
    float* ctx = Vh + HEADSZ;                                   // [B,S,1024]

    const dim3 blk(256);

    // 1) QKV projections (z = which matrix); head = blockIdx.x
    proj_kernel<<<dim3(D_MODEL / 64, (BB * SQ) / 128, 3), blk, 0, stream>>>(
        Qin, Kin, Vin, WQ, WK, WV, Qh, Kh, Vh);

    // 2) scaled masked scores -> attn region of d_out (LDS-staged, async DMA)
    scores_kernel<<<dim3(SQ / 64, SQ / 128, BB * NHEAD), blk, 0, stream>>>(
        Qh, Kh, mask, attn);

    // 3) softmax rows in place
    softmax_kernel<<<dim3(BB * NHEAD * SQ), blk, 0, stream>>>(attn);

    // 4) context = attn @ V
    av_kernel<<<dim3(1, SQ / 128, BB * NHEAD), blk, 0, stream>>>(attn, Vh, ctx);

    // 5) output projection + residual
    outproj_kernel<<<dim3(D_MODEL / 64, (BB * SQ) / 128), blk, 0, stream>>>(
        ctx, Wfc, Qin, outq);

    // 6) LayerNorm
    ln_kernel<<<dim3(BB * SQ), blk, 0, stream>>>(outq, gam, bet);
}